// GenAggSparse_36361193128014
// MI455X (gfx1250) — compile-verified
//
#include <hip/hip_runtime.h>
#include <hip/hip_bf16.h>
#include <stdint.h>

#define N_ROWS   4194304
#define D_FEAT   64
#define N_SEG    131072

#define NB1      1024     // pass-1 blocks
#define PASS1_T  256

#define NB2      4096     // pass-2 blocks
#define WAVES_PB 8        // waves per block (wave32)
#define ROWS_PW  128      // rows per wave
#define RB       16       // rows per async batch (16*256B = 4KB)
#define NBATCH   (ROWS_PW / RB)

static __device__ __forceinline__ float clamp199(float v) {
    return fminf(fmaxf(v, -1.99f), 1.99f);
}
static __device__ __forceinline__ float p_eff_from(const float* p) {
    return tanf(clamp199(p[0]) * 0.78539816339744830962f); // pi/4
}

// ---------------------------------------------------------------- zero
__global__ void zero_f32(float* __restrict__ p, int n) {
    int i = blockIdx.x * blockDim.x + threadIdx.x;
    int stride = gridDim.x * blockDim.x;
    for (; i < n; i += stride) p[i] = 0.0f;
}

// ---------------------------------------------------------------- pass 1: column min/max
__global__ __launch_bounds__(PASS1_T) void pass1_minmax(const float* __restrict__ x,
                                                        float* __restrict__ ws) {
    __shared__ float smin[PASS1_T][4];
    __shared__ float smax[PASS1_T][4];
    const float4* x4 = (const float4*)x;
    const int tid = threadIdx.x;
    long long g = (long long)blockIdx.x * PASS1_T + tid;
    const long long stride = (long long)NB1 * PASS1_T;
    const long long n4 = (long long)N_ROWS * D_FEAT / 4;
    // Column of each float4 component is invariant across the grid-stride
    // (stride*4 is a multiple of 64), so per-thread min/max stay per-column.
    float mn0 =  INFINITY, mn1 =  INFINITY, mn2 =  INFINITY, mn3 =  INFINITY;
    float mx0 = -INFINITY, mx1 = -INFINITY, mx2 = -INFINITY, mx3 = -INFINITY;
    for (; g < n4; g += stride) {
        float4 v = x4[g];
        mn0 = fminf(mn0, v.x); mx0 = fmaxf(mx0, v.x);
        mn1 = fminf(mn1, v.y); mx1 = fmaxf(mx1, v.y);
        mn2 = fminf(mn2, v.z); mx2 = fmaxf(mx2, v.z);
        mn3 = fminf(mn3, v.w); mx3 = fmaxf(mx3, v.w);
    }
    smin[tid][0] = mn0; smin[tid][1] = mn1; smin[tid][2] = mn2; smin[tid][3] = mn3;
    smax[tid][0] = mx0; smax[tid][1] = mx1; smax[tid][2] = mx2; smax[tid][3] = mx3;
    __syncthreads();
    if (tid < 64) {
        // thread t covers columns (4t)%64 + j  ->  column d lives in threads
        // t = (d>>2) + 16k, component j = d&3
        const int j = tid & 3, base = tid >> 2;
        float mn = INFINITY, mx = -INFINITY;
#pragma unroll
        for (int k = 0; k < 16; k++) {
            mn = fminf(mn, smin[base + 16 * k][j]);
            mx = fmaxf(mx, smax[base + 16 * k][j]);
        }
        ws[(size_t)blockIdx.x * 64 + tid] = mn;
        ws[(size_t)NB1 * 64 + (size_t)blockIdx.x * 64 + tid] = mx;
    }
}

// ---------------------------------------------------------------- combine -> shifts[64], c2[64]
__global__ __launch_bounds__(256) void combine_stats(const float* __restrict__ wsin,
                                                     float* __restrict__ stats,
                                                     const float* __restrict__ p) {
    __shared__ float smn[4][64], smx[4][64];
    const int tid = threadIdx.x, d = tid & 63, grp = tid >> 6;
    float mn = INFINITY, mx = -INFINITY;
    for (int k = grp; k < NB1; k += 4) {
        mn = fminf(mn, wsin[(size_t)k * 64 + d]);
        mx = fmaxf(mx, wsin[(size_t)NB1 * 64 + (size_t)k * 64 + d]);
    }
    smn[grp][d] = mn; smx[grp][d] = mx;
    __syncthreads();
    if (tid < 64) {
        float m = fminf(fminf(smn[0][d], smn[1][d]), fminf(smn[2][d], smn[3][d]));
        float M = fmaxf(fmaxf(smx[0][d], smx[1][d]), fmaxf(smx[2][d], smx[3][d]));
        const float shift = m - 0.001f;
        const float mnxs = m - shift;   // ~= 0.001 (exact fp value)
        const float mxxs = M - shift;
        const float pe = p_eff_from(p);
        // base-2 midrange shift: c2 = p * (log2(min_xs)+log2(max_xs))/2  (sign-safe)
        const float c2 = pe * 0.5f * (__log2f(mnxs) + __log2f(mxxs));
        stats[d] = shift;
        stats[64 + d] = c2;
    }
}

// ---------------------------------------------------------------- pass 2 helpers
static __device__ __forceinline__ void flush_seg(int s, float a0, float a1, int runN,
                                                 int firstSeg, int lastSeg,
                                                 float* __restrict__ out,
                                                 int* __restrict__ Ncnt, int lane) {
    float* o = out + (size_t)s * 64 + lane;
    if (s > firstSeg && s < lastSeg) {
        // segment strictly interior to this wave's sorted run: exclusively owned
        o[0] = a0; o[32] = a1;
        if (lane == 0) Ncnt[s] = runN;
    } else {
        atomicAdd(o, a0); atomicAdd(o + 32, a1);
        if (lane == 0) atomicAdd(&Ncnt[s], runN);
    }
}

static __device__ __forceinline__ void issue_batch(uint32_t ldsBase, unsigned long long gBase,
                                                   int lane) {
#pragma unroll
    for (int j = 0; j < 8; j++) {
        const uint32_t off = (uint32_t)((j * 32 + lane) * 16);
        uint32_t l = ldsBase + off;
        unsigned long long g = gBase + off;
        asm volatile("global_load_async_to_lds_b128 %0, %1, off"
                     :: "v"(l), "v"(g) : "memory");
    }
}

// ---------------------------------------------------------------- pass 2: segment sums
__global__ __launch_bounds__(256) void pass2_accum(const float* __restrict__ x,
                                                   const long long* __restrict__ index,
                                                   const float* __restrict__ p,
                                                   const float* __restrict__ stats,
                                                   float* __restrict__ out,
                                                   int* __restrict__ Ncnt) {
    __shared__ float smem[WAVES_PB][2][RB * 64];   // 64 KB: per-wave double buffer
    const int tid = threadIdx.x, lane = tid & 31, wave = tid >> 5;
    const long long R0 = ((long long)blockIdx.x * WAVES_PB + wave) * (long long)ROWS_PW;

    const float pe  = p_eff_from(p);
    const float sh0 = stats[lane],      sh1 = stats[lane + 32];
    const float c20 = stats[64 + lane], c21 = stats[96 + lane];

    const uint32_t* idxlo = (const uint32_t*)index;   // little-endian low dwords of int64
    const int firstSeg = (int)idxlo[2 * R0];
    const int lastSeg  = (int)idxlo[2 * (R0 + ROWS_PW - 1)];

    const uint32_t lds0 = (uint32_t)(uintptr_t)&smem[wave][0][0];
    const uint32_t lds1 = (uint32_t)(uintptr_t)&smem[wave][1][0];
    const unsigned long long gbase =
        (unsigned long long)(uintptr_t)x + (unsigned long long)R0 * (D_FEAT * 4);

    issue_batch(lds0, gbase, lane);

    int scur = firstSeg, runN = 0;
    float acc0 = 0.0f, acc1 = 0.0f;

    for (int b = 0; b < NBATCH; b++) {
        if (b + 1 < NBATCH) {
            issue_batch((b + 1) & 1 ? lds1 : lds0,
                        gbase + (unsigned long long)(b + 1) * RB * 256, lane);
            if (b + 2 < NBATCH)
                __builtin_prefetch(&idxlo[2 * (R0 + (long long)(b + 2) * RB)], 0, 0);
            __builtin_amdgcn_s_wait_asynccnt(8);   // batch b complete (in-order)
        } else {
            __builtin_amdgcn_s_wait_asynccnt(0);
        }
        asm volatile("" ::: "memory");

        const int myidx = (int)idxlo[2 * (R0 + (long long)b * RB + (lane & (RB - 1)))];
        const float* lbuf = (b & 1) ? &smem[wave][1][0] : &smem[wave][0][0];
#pragma unroll
        for (int k = 0; k < RB; k++) {
            const int s = __shfl(myidx, k, 32);          // uniform per wave
            const float x0 = lbuf[k * 64 + lane];
            const float x1 = lbuf[k * 64 + 32 + lane];
            // e = 2^(p*log2(x-shift) - c2)  -> v_log_f32 / fma / v_exp_f32
            const float e0 = exp2f(fmaf(pe, __log2f(x0 - sh0), -c20));
            const float e1 = exp2f(fmaf(pe, __log2f(x1 - sh1), -c21));
            if (s != scur) {
                flush_seg(scur, acc0, acc1, runN, firstSeg, lastSeg, out, Ncnt, lane);
                scur = s; acc0 = 0.0f; acc1 = 0.0f; runN = 0;
            }
            acc0 += e0; acc1 += e1; runN++;
        }
    }
    flush_seg(scur, acc0, acc1, runN, firstSeg, lastSeg, out, Ncnt, lane);
}

// ---------------------------------------------------------------- finalize (in place)
__global__ __launch_bounds__(256) void finalize_k(float* __restrict__ out,
                                                  const int* __restrict__ Ncnt,
                                                  const float* __restrict__ stats,
                                                  const float* __restrict__ p,
                                                  const float* __restrict__ a) {
    const int tid = threadIdx.x, d = tid & 63;
    const int s = blockIdx.x * 4 + (tid >> 6);
    const float pe = p_eff_from(p);
    const float inv_p = 1.0f / pe;
    const float av = a[0];
    const float Nv = (float)Ncnt[s];
    const float l2N = __log2f(Nv);                 // N==0 -> -inf -> NaN like reference
    const size_t o = (size_t)s * 64 + d;
    const float sum = out[o];
    const float Y = exp2f(inv_p * (stats[64 + d] + __log2f(sum) - l2N));
    out[o] = exp2f(av * l2N) * (Y + stats[d]);     // N^a * (Y + shift)
}

// ---------------------------------------------------------------- launch
extern "C" void kernel_launch(void* const* d_in, const int* in_sizes, int n_in,
                              void* d_out, int out_size, void* d_ws, size_t ws_size,
                              hipStream_t stream) {
    const float* x          = (const float*)d_in[0];
    const float* p          = (const float*)d_in[1];
    const float* a          = (const float*)d_in[2];
    const long long* index  = (const long long*)d_in[3];
    float* out = (float*)d_out;

    float* wsf   = (float*)d_ws;
    float* stats = wsf + (size_t)2 * NB1 * 64;            // shifts[64] | c2[64]
    int*   Ncnt  = (int*)(wsf + (size_t)2 * NB1 * 64 + 128);

    zero_f32<<<2048, 256, 0, stream>>>(out, N_SEG * D_FEAT);
    zero_f32<<<256, 256, 0, stream>>>((float*)Ncnt, N_SEG);
    pass1_minmax<<<NB1, PASS1_T, 0, stream>>>(x, wsf);
    combine_stats<<<1, 256, 0, stream>>>(wsf, stats, p);
    pass2_accum<<<NB2, 256, 0, stream>>>(x, index, p, stats, out, Ncnt);
    finalize_k<<<N_SEG / 4, 256, 0, stream>>>(out, Ncnt, stats, p, a);
}